// PeriodSparseSelfAttention_83992380440987
// MI455X (gfx1250) — compile-verified
//
#include <hip/hip_runtime.h>

#define DMODEL 512
#define NHEADS 8
#define HD     64
#define BATCH  4
#define SEQ    2048
#define ROWS   (BATCH * SEQ)   /* 8192 */
#define QKVN   (3 * DMODEL)    /* 1536 */

typedef float v2f __attribute__((ext_vector_type(2)));
typedef float v8f __attribute__((ext_vector_type(8)));

__device__ __forceinline__ v2f load_v2(const float* p) { return *(const v2f*)p; }

// ---------------------------------------------------------------------------
// WMMA f32 16x16x4 fragment layouts (CDNA5 ISA §7.12.2):
//  A (16x4):  lane L -> m = L%16 ; VGPR j -> k = 2*(L/16) + j
//  B (4x16):  lane L -> n = L%16 ; VGPR j -> k = 2*(L/16) + j
//  C (16x16): lane L, VGPR r -> m = r + 8*(L/16), n = L%16
// Computes a 32x32 wave tile of C = A(MxK, row-major) * W^T (W is NxK row-major)
// ---------------------------------------------------------------------------
template <int K>
__device__ __forceinline__ void gemm_wave_tile_f32(const float* __restrict__ A,
                                                   const float* __restrict__ W,
                                                   int mBase, int nBase,
                                                   v8f c[2][2]) {
  const int lane = threadIdx.x & 31;
  const int lm   = lane & 15;
  const int kh   = (lane >> 4) << 1;  // 0 or 2

  const float* a0p = A + (size_t)(mBase + lm) * K + kh;
  const float* a1p = A + (size_t)(mBase + 16 + lm) * K + kh;
  const float* b0p = W + (size_t)(nBase + lm) * K + kh;
  const float* b1p = W + (size_t)(nBase + 16 + lm) * K + kh;

#pragma unroll 4
  for (int k0 = 0; k0 < K; k0 += 4) {
    v2f a0 = load_v2(a0p + k0);
    v2f a1 = load_v2(a1p + k0);
    v2f b0 = load_v2(b0p + k0);
    v2f b1 = load_v2(b1p + k0);
    c[0][0] = __builtin_amdgcn_wmma_f32_16x16x4_f32(false, a0, false, b0, (short)0, c[0][0], false, false);
    c[0][1] = __builtin_amdgcn_wmma_f32_16x16x4_f32(false, a0, false, b1, (short)0, c[0][1], false, false);
    c[1][0] = __builtin_amdgcn_wmma_f32_16x16x4_f32(false, a1, false, b0, (short)0, c[1][0], false, false);
    c[1][1] = __builtin_amdgcn_wmma_f32_16x16x4_f32(false, a1, false, b1, (short)0, c[1][1], false, false);
  }
}

// ---------------------------------------------------------------------------
// Kernel 1: qkv = x @ in_proj_w^T + in_proj_b, scattered into head-major
// q/k/v buffers shaped (B, H, T, HD).
// grid = (ROWS/64, QKVN/64), block = 128 (4 waves, 2x2 arrangement of 32x32)
// ---------------------------------------------------------------------------
__global__ void __launch_bounds__(128) qkv_proj_kernel(
    const float* __restrict__ x, const float* __restrict__ w,
    const float* __restrict__ bias,
    float* __restrict__ q, float* __restrict__ k, float* __restrict__ v) {
  const int wave  = threadIdx.x >> 5;
  const int lane  = threadIdx.x & 31;
  const int mBase = blockIdx.x * 64 + (wave >> 1) * 32;
  const int nBase = blockIdx.y * 64 + (wave & 1) * 32;

  const v8f z = {0.f, 0.f, 0.f, 0.f, 0.f, 0.f, 0.f, 0.f};
  v8f c[2][2] = {{z, z}, {z, z}};

  gemm_wave_tile_f32<DMODEL>(x, w, mBase, nBase, c);

  const int lm = lane & 15;
  const int mh = (lane >> 4) << 3;  // 0 or 8
  float* outs[3] = {q, k, v};

#pragma unroll
  for (int i = 0; i < 2; ++i) {
#pragma unroll
    for (int j = 0; j < 2; ++j) {
      const int n   = nBase + j * 16 + lm;
      const float bv = bias[n];
      const int sec = n >> 9;           // /512 : 0=q 1=k 2=v
      const int rem = n & (DMODEL - 1);
      const int h   = rem >> 6;
      const int d   = rem & (HD - 1);
      float* dst = outs[sec];
#pragma unroll
      for (int r = 0; r < 8; ++r) {
        const int m = mBase + i * 16 + mh + r;
        const int b = m >> 11;          // /2048
        const int t = m & (SEQ - 1);
        dst[(((size_t)(b * NHEADS + h) * SEQ) + t) * HD + d] = c[i][j][r] + bv;
      }
    }
  }
}

// ---------------------------------------------------------------------------
// Kernel 2: period-sparse streaming-softmax attention.
// One wave32 per (b,h,t): lanes hold 2 of 64 head dims. Keys j = t, t-P, ...
// grid = B*H*T/8, block = 256 (8 waves)
// ---------------------------------------------------------------------------
__global__ void __launch_bounds__(256) sparse_attn_kernel(
    const float* __restrict__ q, const float* __restrict__ k,
    const float* __restrict__ v, const int* __restrict__ periods,
    float* __restrict__ o) {
  const int wid  = (blockIdx.x << 3) + (threadIdx.x >> 5);
  const int lane = threadIdx.x & 31;
  const int t  = wid & (SEQ - 1);
  const int bh = wid >> 11;             // b*H + h
  const int b  = bh >> 3;
  const int h  = bh & (NHEADS - 1);

  int P = periods[bh];
  if (P < 1) P = 1;

  const float* kbase = k + (size_t)bh * SEQ * HD;
  const float* vbase = v + (size_t)bh * SEQ * HD;

  v2f qv = load_v2(q + ((size_t)bh * SEQ + t) * HD + lane * 2);
  qv.x *= 0.125f;  // 1/sqrt(64)
  qv.y *= 0.125f;

  float m  = -__builtin_inff();
  float s  = 0.f;
  float o0 = 0.f, o1 = 0.f;

  for (int j = t; j >= 0; j -= P) {
    v2f kv = load_v2(kbase + (size_t)j * HD + lane * 2);
    float d = qv.x * kv.x + qv.y * kv.y;
    d += __shfl_xor(d, 16, 32);
    d += __shfl_xor(d, 8, 32);
    d += __shfl_xor(d, 4, 32);
    d += __shfl_xor(d, 2, 32);
    d += __shfl_xor(d, 1, 32);

    const float nm    = fmaxf(m, d);
    const float scale = __expf(m - nm);   // 0 on first iter (m = -inf)
    const float p     = __expf(d - nm);
    v2f vv = load_v2(vbase + (size_t)j * HD + lane * 2);
    s  = s * scale + p;
    o0 = o0 * scale + p * vv.x;
    o1 = o1 * scale + p * vv.y;
    m  = nm;
  }

  const float inv = 1.f / s;
  float* op = o + ((size_t)b * SEQ + t) * DMODEL + h * HD + lane * 2;
  op[0] = o0 * inv;
  op[1] = o1 * inv;
}

// ---------------------------------------------------------------------------
// Kernel 3: final = attn_out @ out_w^T + out_b   (8192x512 @ 512x512)
// grid = (ROWS/64, DMODEL/64), block = 128
// ---------------------------------------------------------------------------
__global__ void __launch_bounds__(128) out_proj_kernel(
    const float* __restrict__ a, const float* __restrict__ w,
    const float* __restrict__ bias, float* __restrict__ out) {
  const int wave  = threadIdx.x >> 5;
  const int lane  = threadIdx.x & 31;
  const int mBase = blockIdx.x * 64 + (wave >> 1) * 32;
  const int nBase = blockIdx.y * 64 + (wave & 1) * 32;

  const v8f z = {0.f, 0.f, 0.f, 0.f, 0.f, 0.f, 0.f, 0.f};
  v8f c[2][2] = {{z, z}, {z, z}};

  gemm_wave_tile_f32<DMODEL>(a, w, mBase, nBase, c);

  const int lm = lane & 15;
  const int mh = (lane >> 4) << 3;
#pragma unroll
  for (int i = 0; i < 2; ++i) {
#pragma unroll
    for (int j = 0; j < 2; ++j) {
      const int n    = nBase + j * 16 + lm;
      const float bv = bias[n];
#pragma unroll
      for (int r = 0; r < 8; ++r) {
        const int m = mBase + i * 16 + mh + r;
        out[(size_t)m * DMODEL + n] = c[i][j][r] + bv;
      }
    }
  }
}

// ---------------------------------------------------------------------------
extern "C" void kernel_launch(void* const* d_in, const int* in_sizes, int n_in,
                              void* d_out, int out_size, void* d_ws, size_t ws_size,
                              hipStream_t stream) {
  const float* x        = (const float*)d_in[0];
  const int*   periods  = (const int*)d_in[1];
  const float* in_w     = (const float*)d_in[2];
  const float* in_b     = (const float*)d_in[3];
  const float* out_w    = (const float*)d_in[4];
  const float* out_b    = (const float*)d_in[5];
  float*       out      = (float*)d_out;

  const size_t nElem = (size_t)BATCH * NHEADS * SEQ * HD;  // 4M floats each
  float* q    = (float*)d_ws;
  float* k    = q + nElem;
  float* v    = k + nElem;
  float* attn = v + nElem;                                  // (B,T,DMODEL)

  {
    dim3 grid(ROWS / 64, QKVN / 64);
    qkv_proj_kernel<<<grid, 128, 0, stream>>>(x, in_w, in_b, q, k, v);
  }
  {
    dim3 grid((BATCH * NHEADS * SEQ) / 8);
    sparse_attn_kernel<<<grid, 256, 0, stream>>>(q, k, v, periods, attn);
  }
  {
    dim3 grid(ROWS / 64, DMODEL / 64);
    out_proj_kernel<<<grid, 128, 0, stream>>>(attn, out_w, out_b, out);
  }
}